// HeteroGAT_71622874628355
// MI455X (gfx1250) — compile-verified
//
#include <hip/hip_runtime.h>
#include <cstddef>

#define NPAPER 100000
#define NAUTHOR 50000
#define NEDGE 400000
#define D_IN 128
#define HID 64
#define OUT_DIM 349
#define CHUNK 64
#define NEG_SLOPE 0.2f
#define CPAD_MAX 352   // OUT_DIM padded to multiple of 16

typedef __attribute__((ext_vector_type(2))) float v2f;
typedef __attribute__((ext_vector_type(8))) float v8f;

// ---- order-preserving float<->uint encoding for atomic segment-max ----
__device__ __forceinline__ unsigned enc_f32(float f) {
    unsigned u = __float_as_uint(f);
    return (u & 0x80000000u) ? ~u : (u | 0x80000000u);
}
__device__ __forceinline__ float dec_f32(unsigned u) {
    return (u & 0x80000000u) ? __uint_as_float(u & 0x7fffffffu)
                             : __uint_as_float(~u);
}
// enc(-inf) == 0x007FFFFF
#define ENC_NEG_INF 0x007FFFFFu

// =====================================================================
// W transpose + zero-pad:  Wt[c*K + k] = (c < Ctot) ? W[k*Ctot + c] : 0
// Makes B fragments contiguous (b64 loads) and removes all bounds
// checks / exec-mask juggling from the GEMM hot loop.
// =====================================================================
__global__ void transpose_pad_w(const float* __restrict__ W, float* __restrict__ Wt,
                                int K, int Ctot, int Cpad) {
    int idx = (int)(blockIdx.x * blockDim.x + threadIdx.x);
    if (idx >= Cpad * K) return;
    int c = idx / K;
    int k = idx - c * K;
    Wt[idx] = (c < Ctot) ? W[(size_t)k * Ctot + c] : 0.f;
}

// =====================================================================
// GEMM: Hc[n, c-c0] = sum_k X[n,k] * Wt^T[k,c]   for c in [c0, c0+cw)
// One wave computes TWO 16x16 M-tiles (shared B fragment) via
// V_WMMA_F32_16X16X4_F32. Hc row stride = CHUNK. K % 4 == 0.
// =====================================================================
__global__ void gemm_wmma_f32(const float* __restrict__ X,
                              const float* __restrict__ Wt,   // [Cpad x K] transposed
                              float* __restrict__ Hc,
                              int N, int K, int Ctot, int c0, int tiles_c) {
    int gw   = (int)((blockIdx.x * blockDim.x + threadIdx.x) >> 5);
    int lane = (int)(threadIdx.x & 31);
    int tiles_n = N >> 4;
    int groups  = (tiles_n + 1) >> 1;           // 2 M-tiles per wave
    if (gw >= groups * tiles_c) return;
    int tc = gw % tiles_c;
    int g  = gw / tiles_c;
    int tn0 = 2 * g;
    int tn1 = 2 * g + 1;

    int half = lane >> 4;                       // 0: lanes 0-15, 1: lanes 16-31
    int lm   = lane & 15;

    int row0 = (tn0 << 4) + lm;
    int row1 = (tn1 << 4) + lm;
    int row1c = (row1 < N) ? row1 : (N - 1);    // clamp (dup loads are harmless)
    int col  = c0 + (tc << 4) + lm;             // always < Cpad

    const float* __restrict__ xr0  = X + (size_t)row0 * K;
    const float* __restrict__ xr1  = X + (size_t)row1c * K;
    const float* __restrict__ wcol = Wt + (size_t)col * K;

    v8f acc0 = {0.f, 0.f, 0.f, 0.f, 0.f, 0.f, 0.f, 0.f};
    v8f acc1 = {0.f, 0.f, 0.f, 0.f, 0.f, 0.f, 0.f, 0.f};
    for (int k0 = 0; k0 < K; k0 += 4) {
        int ka = k0 + 2 * half;                 // K pair handled by this half-wave
        v2f b;
        b[0] = wcol[ka];
        b[1] = wcol[ka + 1];
        v2f a0;
        a0[0] = xr0[ka];
        a0[1] = xr0[ka + 1];
        v2f a1;
        a1[0] = xr1[ka];
        a1[1] = xr1[ka + 1];
        acc0 = __builtin_amdgcn_wmma_f32_16x16x4_f32(
            false, a0, false, b, (short)0, acc0, false, false);
        acc1 = __builtin_amdgcn_wmma_f32_16x16x4_f32(
            false, a1, false, b, (short)0, acc1, false, false);
    }

    // C/D layout: VGPR r, lanes 0-15 -> M = r, lanes 16-31 -> M = r + 8
    bool cok = (col < Ctot);
    int ccol = (tc << 4) + lm;                  // chunk-local column
    if (cok) {
        float* o0 = Hc + (size_t)((tn0 << 4) + 8 * half) * CHUNK + ccol;
#pragma unroll
        for (int r = 0; r < 8; ++r)
            o0[(size_t)r * CHUNK] = acc0[r];
        if (tn1 < tiles_n) {
            float* o1 = Hc + (size_t)((tn1 << 4) + 8 * half) * CHUNK + ccol;
#pragma unroll
            for (int r = 0; r < 8; ++r)
                o1[(size_t)r * CHUNK] = acc1[r];
        }
    }
}

// =====================================================================
// u[d] = sum_h W[d,h] * a[h]     (tiny: d < K <= 128, h < H <= 349)
// =====================================================================
__global__ void matvec_w(const float* __restrict__ W, const float* __restrict__ av,
                         float* __restrict__ u, int K, int H) {
    int d = (int)(blockIdx.x * blockDim.x + threadIdx.x);
    if (d >= K) return;
    float s = 0.f;
    for (int h = 0; h < H; ++h) s += W[(size_t)d * H + h] * av[h];
    u[d] = s;
}

// =====================================================================
// a[n] = sum_k X[n,k] * u[k]   — one wave per node, shuffle reduce
// =====================================================================
__global__ void matvec_nodes(const float* __restrict__ X, const float* __restrict__ u,
                             float* __restrict__ a, int N, int K) {
    int w    = (int)((blockIdx.x * blockDim.x + threadIdx.x) >> 5);
    int lane = (int)(threadIdx.x & 31);
    if (w >= N) return;
    const float* xr = X + (size_t)w * K;
    float s = 0.f;
    for (int k = lane; k < K; k += 32) s += xr[k] * u[k];
    for (int off = 16; off > 0; off >>= 1) s += __shfl_xor(s, off, 32);
    if (lane == 0) a[w] = s;
}

// =====================================================================
// init kernels
// =====================================================================
__global__ void init_softmax(unsigned* __restrict__ menc, float* __restrict__ denom, int N) {
    int i = (int)(blockIdx.x * blockDim.x + threadIdx.x);
    if (i < N) { menc[i] = ENC_NEG_INF; denom[i] = 0.f; }
}

__global__ void init_out_bias(float* __restrict__ out, const float* __restrict__ b0,
                              const float* __restrict__ b1, int N, int C) {
    size_t i = (size_t)blockIdx.x * blockDim.x + threadIdx.x;
    size_t total = (size_t)N * C;
    if (i >= total) return;
    int c = (int)(i % C);
    float v = b0[c];
    if (b1) v += b1[c];
    out[i] = v;
}

// =====================================================================
// edge passes
// =====================================================================
__global__ void edge_logits(const int* __restrict__ src, const int* __restrict__ dst,
                            const float* __restrict__ as, const float* __restrict__ ad,
                            float* __restrict__ ebuf, unsigned* __restrict__ menc, int E) {
    int i = (int)(blockIdx.x * blockDim.x + threadIdx.x);
    if (i >= E) return;
    int d = dst[i];
    float x = as[src[i]] + ad[d];
    x = (x >= 0.f) ? x : NEG_SLOPE * x;        // leaky_relu
    ebuf[i] = x;
    atomicMax(&menc[d], enc_f32(x));
}

__global__ void edge_exp(const int* __restrict__ dst, float* __restrict__ ebuf,
                         const unsigned* __restrict__ menc, float* __restrict__ denom, int E) {
    int i = (int)(blockIdx.x * blockDim.x + threadIdx.x);
    if (i >= E) return;
    int d = dst[i];
    float ex = expf(ebuf[i] - dec_f32(menc[d]));
    ebuf[i] = ex;
    atomicAdd(&denom[d], ex);
}

__global__ void edge_norm(const int* __restrict__ dst, float* __restrict__ ebuf,
                          const float* __restrict__ denom, int E) {
    int i = (int)(blockIdx.x * blockDim.x + threadIdx.x);
    if (i >= E) return;
    ebuf[i] = ebuf[i] / fmaxf(denom[dst[i]], 1e-16f);
}

// out[dst, c0+c] += alpha[e] * Hc[src, c]   — one wave per edge, lanes over cols
__global__ void edge_agg(const int* __restrict__ src, const int* __restrict__ dst,
                         const float* __restrict__ alpha, const float* __restrict__ Hc,
                         float* __restrict__ out, int E, int cw, int c0, int ldout) {
    int w    = (int)((blockIdx.x * blockDim.x + threadIdx.x) >> 5);
    int lane = (int)(threadIdx.x & 31);
    if (w >= E) return;
    int s = src[w], d = dst[w];
    float al = alpha[w];
    const float* hr = Hc + (size_t)s * CHUNK;
    float* orow = out + (size_t)d * ldout + c0;
    for (int c = lane; c < cw; c += 32)
        atomicAdd(&orow[c], al * hr[c]);
}

// =====================================================================
// elementwise ELU (alpha = 1)
// =====================================================================
__global__ void elu_k(float* __restrict__ x, size_t n) {
    size_t i = (size_t)blockIdx.x * blockDim.x + threadIdx.x;
    if (i >= n) return;
    float v = x[i];
    x[i] = (v > 0.f) ? v : expm1f(v);
}

// =====================================================================
// host-side orchestration
// =====================================================================
namespace {

struct Ws {
    float*    hsC;     // NPAPER * CHUNK
    float*    alpha;   // NEDGE
    float*    a_s;     // NPAPER
    float*    a_d;     // NPAPER
    unsigned* menc;    // NPAPER
    float*    denom;   // NPAPER
    float*    u;       // 128
    float*    v;       // 128
    float*    Wt;      // CPAD_MAX * D_IN  (transposed, zero-padded weights)
    float*    hp;      // NPAPER * HID
    float*    ha;      // NAUTHOR * HID
};

inline int cdiv(int a, int b) { return (a + b - 1) / b; }

void run_gat(hipStream_t st, const Ws& w,
             const float* x_src, const float* x_dst, int Ns, int Nd, int Din,
             const int* src, const int* dst, int E,
             const float* W_dst, const float* W_src,
             const float* a_dst, const float* a_src,
             int H, float* out, int ldout) {
    // attention scalar projections:  u = W_src @ a_src,  v = W_dst @ a_dst
    matvec_w<<<cdiv(Din, 128), 128, 0, st>>>(W_src, a_src, w.u, Din, H);
    matvec_w<<<cdiv(Din, 128), 128, 0, st>>>(W_dst, a_dst, w.v, Din, H);
    matvec_nodes<<<cdiv(Ns, 8), 256, 0, st>>>(x_src, w.u, w.a_s, Ns, Din);
    matvec_nodes<<<cdiv(Nd, 8), 256, 0, st>>>(x_dst, w.v, w.a_d, Nd, Din);

    // edge softmax over destination segments
    init_softmax<<<cdiv(Nd, 256), 256, 0, st>>>(w.menc, w.denom, Nd);
    edge_logits<<<cdiv(E, 256), 256, 0, st>>>(src, dst, w.a_s, w.a_d, w.alpha, w.menc, E);
    edge_exp<<<cdiv(E, 256), 256, 0, st>>>(dst, w.alpha, w.menc, w.denom, E);
    edge_norm<<<cdiv(E, 256), 256, 0, st>>>(dst, w.alpha, w.denom, E);

    // transposed / padded weights for the GEMM
    int Cpad = cdiv(H, 16) * 16;
    transpose_pad_w<<<cdiv(Cpad * Din, 256), 256, 0, st>>>(W_src, w.Wt, Din, H, Cpad);

    // column-chunked:  hs chunk via WMMA GEMM, then weighted scatter-add
    int tiles_n = Ns >> 4;
    int groups  = (tiles_n + 1) >> 1;
    for (int c0 = 0; c0 < H; c0 += CHUNK) {
        int cw = (H - c0 < CHUNK) ? (H - c0) : CHUNK;
        int tiles_c = cdiv(cw, 16);
        int waves = groups * tiles_c;
        gemm_wmma_f32<<<cdiv(waves, 8), 256, 0, st>>>(x_src, w.Wt, w.hsC,
                                                      Ns, Din, H, c0, tiles_c);
        edge_agg<<<cdiv(E, 8), 256, 0, st>>>(src, dst, w.alpha, w.hsC,
                                             out, E, cw, c0, ldout);
    }
}

} // namespace

extern "C" void kernel_launch(void* const* d_in, const int* in_sizes, int n_in,
                              void* d_out, int out_size, void* d_ws, size_t ws_size,
                              hipStream_t stream) {
    (void)in_sizes; (void)n_in; (void)out_size; (void)ws_size;

    const float* x_p  = (const float*)d_in[0];
    const float* x_a  = (const float*)d_in[1];
    const int* ci_s   = (const int*)d_in[2];
    const int* ci_d   = (const int*)d_in[3];
    const int* wr_s   = (const int*)d_in[4];
    const int* wr_d   = (const int*)d_in[5];
    const int* wb_s   = (const int*)d_in[6];
    const int* wb_d   = (const int*)d_in[7];

    // JAX pytree flatten order: dict keys sorted.
    // layers: l1 < l2 ; edge types: cites < wby < writes ;
    // per-conv leaves: W_dst < W_src < a_dst < a_src < b
    const float* P[30];
    for (int i = 0; i < 30; ++i) P[i] = (const float*)d_in[8 + i];
    // l1: cites P[0..4], wby P[5..9], writes P[10..14]
    // l2: cites P[15..19], wby P[20..24], writes P[25..29]

    // workspace carve
    Ws w;
    {
        float* b = (float*)d_ws;
        size_t o = 0;
        w.hsC   = b + o; o += (size_t)NPAPER * CHUNK;
        w.alpha = b + o; o += NEDGE;
        w.a_s   = b + o; o += NPAPER;
        w.a_d   = b + o; o += NPAPER;
        w.menc  = (unsigned*)(b + o); o += NPAPER;
        w.denom = b + o; o += NPAPER;
        w.u     = b + o; o += 128;
        w.v     = b + o; o += 128;
        w.Wt    = b + o; o += (size_t)CPAD_MAX * D_IN;
        w.hp    = b + o; o += (size_t)NPAPER * HID;
        w.ha    = b + o; o += (size_t)NAUTHOR * HID;
    }

    // ---------------- Layer 1 ----------------
    // out_p1 = bias(cites) + bias(writes) + aggregations ; out_a1 = bias(wby) + agg
    init_out_bias<<<cdiv((int)((size_t)NPAPER * HID), 256), 256, 0, stream>>>(
        w.hp, P[4], P[14], NPAPER, HID);
    init_out_bias<<<cdiv((int)((size_t)NAUTHOR * HID), 256), 256, 0, stream>>>(
        w.ha, P[9], nullptr, NAUTHOR, HID);

    // cites: paper -> paper
    run_gat(stream, w, x_p, x_p, NPAPER, NPAPER, D_IN, ci_s, ci_d, NEDGE,
            P[0], P[1], P[2], P[3], HID, w.hp, HID);
    // writes: author -> paper
    run_gat(stream, w, x_a, x_p, NAUTHOR, NPAPER, D_IN, wr_s, wr_d, NEDGE,
            P[10], P[11], P[12], P[13], HID, w.hp, HID);
    // wby: paper -> author
    run_gat(stream, w, x_p, x_a, NPAPER, NAUTHOR, D_IN, wb_s, wb_d, NEDGE,
            P[5], P[6], P[7], P[8], HID, w.ha, HID);

    elu_k<<<cdiv((int)((size_t)NPAPER * HID), 256), 256, 0, stream>>>(w.hp, (size_t)NPAPER * HID);
    elu_k<<<cdiv((int)((size_t)NAUTHOR * HID), 256), 256, 0, stream>>>(w.ha, (size_t)NAUTHOR * HID);

    // ---------------- Layer 2 (writes straight into d_out) ----------------
    float* out_p = (float*)d_out;
    float* out_a = out_p + (size_t)NPAPER * OUT_DIM;

    init_out_bias<<<cdiv((int)((size_t)NPAPER * OUT_DIM), 256), 256, 0, stream>>>(
        out_p, P[19], P[29], NPAPER, OUT_DIM);
    init_out_bias<<<cdiv((int)((size_t)NAUTHOR * OUT_DIM), 256), 256, 0, stream>>>(
        out_a, P[24], nullptr, NAUTHOR, OUT_DIM);

    // cites: paper -> paper
    run_gat(stream, w, w.hp, w.hp, NPAPER, NPAPER, HID, ci_s, ci_d, NEDGE,
            P[15], P[16], P[17], P[18], OUT_DIM, out_p, OUT_DIM);
    // writes: author -> paper
    run_gat(stream, w, w.ha, w.hp, NAUTHOR, NPAPER, HID, wr_s, wr_d, NEDGE,
            P[25], P[26], P[27], P[28], OUT_DIM, out_p, OUT_DIM);
    // wby: paper -> author
    run_gat(stream, w, w.hp, w.ha, NPAPER, NAUTHOR, HID, wb_s, wb_d, NEDGE,
            P[20], P[21], P[22], P[23], OUT_DIM, out_a, OUT_DIM);
}